// Our_44744969290484
// MI455X (gfx1250) — compile-verified
//
#include <hip/hip_runtime.h>

#define USER_N 100000
#define ITEM_N 200000
#define NN     (USER_N + ITEM_N)
#define LATDIM 128
#define NE     600000
#define UE     300000

typedef float v2f __attribute__((ext_vector_type(2)));
typedef float v8f __attribute__((ext_vector_type(8)));

__device__ __forceinline__ int rfl_i(int v) {
    return __builtin_amdgcn_readfirstlane(v);
}
__device__ __forceinline__ float rfl_f(float v) {
    return __int_as_float(__builtin_amdgcn_readfirstlane(__float_as_int(v)));
}

// ---------------- init kernels ----------------

__global__ void k_init_concat(const float4* __restrict__ u, const float4* __restrict__ it,
                              float4* __restrict__ cur, float4* __restrict__ acc,
                              int total4, int user4) {
    int i = blockIdx.x * blockDim.x + threadIdx.x;
    if (i >= total4) return;
    float4 v = (i < user4) ? u[i] : it[i - user4];
    cur[i] = v;
    acc[i] = v;
}

__global__ void k_copy2(const float4* __restrict__ src, float4* __restrict__ a,
                        float4* __restrict__ b, int n4) {
    int i = blockIdx.x * blockDim.x + threadIdx.x;
    if (i >= n4) return;
    float4 v = src[i];
    a[i] = v;
    b[i] = v;
}

__global__ void k_zero_i32(int* __restrict__ p, int n) {
    int i = blockIdx.x * blockDim.x + threadIdx.x;
    if (i < n) p[i] = 0;
}

// ---------------- CSR build ----------------

__global__ void k_count(const int* __restrict__ rows, int* __restrict__ cnt, int E) {
    int i = blockIdx.x * blockDim.x + threadIdx.x;
    if (i < E) atomicAdd(&cnt[rows[i]], 1);
}

// Single-workgroup exclusive scan: cnt[0..n-1] -> ptr[0..n] and cursor[0..n-1].
// cnt may alias cursor (each thread reads its own element before overwriting it).
__global__ void __launch_bounds__(1024) k_scan(const int* __restrict__ cnt,
                                               int* __restrict__ ptr,
                                               int* __restrict__ cursor, int n) {
    __shared__ int sdata[1024];
    __shared__ int soff;
    if (threadIdx.x == 0) soff = 0;
    __syncthreads();
    for (int base = 0; base < n; base += 1024) {
        int i = base + (int)threadIdx.x;
        int v = (i < n) ? cnt[i] : 0;
        sdata[threadIdx.x] = v;
        __syncthreads();
        for (int d = 1; d < 1024; d <<= 1) {
            int t = (threadIdx.x >= (unsigned)d) ? sdata[threadIdx.x - d] : 0;
            __syncthreads();
            sdata[threadIdx.x] += t;
            __syncthreads();
        }
        int incl = sdata[threadIdx.x];
        int off  = soff;
        if (i < n) {
            int ex = off + incl - v;
            ptr[i]    = ex;
            cursor[i] = ex;
        }
        __syncthreads();
        if (threadIdx.x == 1023) soff = off + sdata[1023];
        __syncthreads();
    }
    if (threadIdx.x == 0) ptr[n] = soff;
}

__global__ void k_scatter(const int* __restrict__ rows, const int* __restrict__ cols,
                          const float* __restrict__ vals, int* __restrict__ cursor,
                          int* __restrict__ ccol, float* __restrict__ cval, int E) {
    int i = blockIdx.x * blockDim.x + threadIdx.x;
    if (i >= E) return;
    int p = atomicAdd(&cursor[rows[i]], 1);
    ccol[p] = cols[i];
    cval[p] = vals[i];
}

// ---------------- WMMA row-blocked SpMM ----------------
// One wave32 per 16-row output tile. Edge metadata is forced scalar via
// readfirstlane: the iteration loop is SALU-controlled (EXEC untouched around
// WMMAs) and the local-row search runs on the scalar pipe. Each iteration
// consumes 8 edges = two K=4 WMMA passes: A(16x4) routes edge values to
// destination rows, B(4x16) is a 16-column slab of the gathered source rows
// (each 16-lane half reads a contiguous 64B segment). B operands are
// materialized in explicit arrays so all 32 gathers issue before the first
// WMMA (staggered s_wait_loadcnt, maximal memory-level parallelism). Rows are
// exclusively owned per tile -> plain stores, no atomics.
__global__ void __launch_bounds__(128) k_spmm_wmma(const int* __restrict__ rp,
                                                   const int* __restrict__ ccol,
                                                   const float* __restrict__ cval,
                                                   const float* __restrict__ x,
                                                   float* __restrict__ nxt,
                                                   float* __restrict__ acc, int n) {
    const int lane = (int)(threadIdx.x & 31u);
    const int wv   = (int)(threadIdx.x >> 5u);
    const int tile = blockIdx.x * 4 + wv;
    const int ntiles = (n + 15) >> 4;
    if (tile >= ntiles) return;              // wave-uniform

    const int  r0 = tile << 4;
    const int  m  = lane & 15;
    const bool hi = lane >= 16;

    // Scalar row boundaries rp[r0+1 .. r0+15] (clamped) in SGPRs.
    int b[15];
#pragma unroll
    for (int j = 0; j < 15; ++j) {
        int r = r0 + 1 + j;
        b[j] = rfl_i(rp[r > n ? n : r]);
    }
    const int start = rfl_i(rp[r0]);
    const int re    = r0 + 16;
    const int end   = rfl_i(rp[re > n ? n : re]);

    const float* xm = x + m;                 // per-lane column base

    v8f c[8] = {};

    for (int base = start; base < end; base += 8) {   // scalar loop control
        int   col[8];
        float vv[8];
        int   lr[8];
#pragma unroll
        for (int k = 0; k < 8; ++k) {
            int e  = base + k;                        // scalar
            int ec = (e < end) ? e : (end - 1);       // scalar clamp, no branch
            col[k] = rfl_i(ccol[ec]);
            float v_ = rfl_f(cval[ec]);
            vv[k]  = (e < end) ? v_ : 0.f;            // scalar select
            int l_ = 0;
#pragma unroll
            for (int j = 0; j < 15; ++j) l_ += (e >= b[j]);   // SALU search
            lr[k] = l_;
        }

        // A operands, 16x4 f32: lanes 0-15 hold K=0 (vgpr0) / K=1 (vgpr1) for
        // M=lane; lanes 16-31 hold K=2 / K=3.
        v2f A0, A1;
        {
            int   sl0 = hi ? lr[2] : lr[0];
            int   sl1 = hi ? lr[3] : lr[1];
            float sv0 = hi ? vv[2] : vv[0];
            float sv1 = hi ? vv[3] : vv[1];
            A0.x = (sl0 == m) ? sv0 : 0.f;
            A0.y = (sl1 == m) ? sv1 : 0.f;
        }
        {
            int   sl0 = hi ? lr[6] : lr[4];
            int   sl1 = hi ? lr[7] : lr[5];
            float sv0 = hi ? vv[6] : vv[4];
            float sv1 = hi ? vv[7] : vv[5];
            A1.x = (sl0 == m) ? sv0 : 0.f;
            A1.y = (sl1 == m) ? sv1 : 0.f;
        }

        // B operands, 4x16 f32 (mirror of A): vgpr0 = rows K=0 (lanes 0-15) /
        // K=2 (lanes 16-31); vgpr1 = K=1 / K=3.
        const float* p00 = xm + (size_t)(hi ? col[2] : col[0]) * LATDIM;
        const float* p01 = xm + (size_t)(hi ? col[3] : col[1]) * LATDIM;
        const float* p10 = xm + (size_t)(hi ? col[6] : col[4]) * LATDIM;
        const float* p11 = xm + (size_t)(hi ? col[7] : col[5]) * LATDIM;

        v2f B0[8], B1[8];
#pragma unroll
        for (int s = 0; s < 8; ++s) {
            B0[s].x = p00[s * 16];
            B0[s].y = p01[s * 16];
        }
#pragma unroll
        for (int s = 0; s < 8; ++s) {
            B1[s].x = p10[s * 16];
            B1[s].y = p11[s * 16];
        }
#pragma unroll
        for (int s = 0; s < 8; ++s)
            c[s] = __builtin_amdgcn_wmma_f32_16x16x4_f32(
                false, A0, false, B0[s], (short)0, c[s], false, false);
#pragma unroll
        for (int s = 0; s < 8; ++s)
            c[s] = __builtin_amdgcn_wmma_f32_16x16x4_f32(
                false, A1, false, B1[s], (short)0, c[s], false, false);
    }

    // C/D 16x16 f32 layout: vgpr v, lanes 0-15 -> (M=v, N=lane),
    // lanes 16-31 -> (M=v+8, N=lane-16).
#pragma unroll
    for (int v = 0; v < 8; ++v) {
        int grow = r0 + v + (hi ? 8 : 0);
        if (grow < n) {
            size_t rowoff = (size_t)grow * LATDIM + m;
#pragma unroll
            for (int s = 0; s < 8; ++s) {
                float d = c[s][v];
                nxt[rowoff + s * 16]  = d;
                acc[rowoff + s * 16] += d;
            }
        }
    }
}

// ---------------- launch ----------------

extern "C" void kernel_launch(void* const* d_in, const int* in_sizes, int n_in,
                              void* d_out, int out_size, void* d_ws, size_t ws_size,
                              hipStream_t stream) {
    const float* uE    = (const float*)d_in[0];
    const float* iE    = (const float*)d_in[1];
    const int*   adj_r = (const int*)d_in[2];
    const int*   adj_c = (const int*)d_in[3];
    const float* adj_v = (const float*)d_in[4];
    const int*   uu_r  = (const int*)d_in[5];
    const int*   uu_c  = (const int*)d_in[6];
    const float* uu_v  = (const float*)d_in[7];

    float* acc  = (float*)d_out;                              // NN x 128
    float* uacc = (float*)d_out + (size_t)NN * LATDIM;        // USER x 128

    // Workspace layout (256B aligned slabs).
    char*  w = (char*)d_ws;
    auto alloc = [&](size_t bytes) {
        char* p = w;
        w += (bytes + 255) & ~(size_t)255;
        return p;
    };
    float* cur   = (float*)alloc((size_t)NN * LATDIM * sizeof(float));
    float* nxt   = (float*)alloc((size_t)NN * LATDIM * sizeof(float));
    int*   aptr  = (int*)alloc((size_t)(NN + 1) * sizeof(int));
    int*   acur  = (int*)alloc((size_t)(NN + 1) * sizeof(int));
    int*   accol = (int*)alloc((size_t)NE * sizeof(int));
    float* acval = (float*)alloc((size_t)NE * sizeof(float));
    int*   uptr  = (int*)alloc((size_t)(USER_N + 1) * sizeof(int));
    int*   ucur  = (int*)alloc((size_t)(USER_N + 1) * sizeof(int));
    int*   uccol = (int*)alloc((size_t)UE * sizeof(int));
    float* ucval = (float*)alloc((size_t)UE * sizeof(float));
    (void)ws_size; (void)in_sizes; (void)n_in; (void)out_size;

    // 1) cur = concat(uE, iE); acc = cur
    {
        int total4 = NN * LATDIM / 4, user4 = USER_N * LATDIM / 4;
        k_init_concat<<<(total4 + 255) / 256, 256, 0, stream>>>(
            (const float4*)uE, (const float4*)iE, (float4*)cur, (float4*)acc,
            total4, user4);
    }

    // 2) CSR for the full adjacency (built once, reused for all 3 layers)
    k_zero_i32<<<(NN + 1 + 255) / 256, 256, 0, stream>>>(acur, NN + 1);
    k_count<<<(NE + 255) / 256, 256, 0, stream>>>(adj_r, acur, NE);
    k_scan<<<1, 1024, 0, stream>>>(acur, aptr, acur, NN);
    k_scatter<<<(NE + 255) / 256, 256, 0, stream>>>(adj_r, adj_c, adj_v, acur,
                                                    accol, acval, NE);

    // 3) 3 GNN layers: nxt = A*cur; acc += nxt; swap
    {
        int ntiles = (NN + 15) / 16;
        int blocks = (ntiles + 3) / 4;
        for (int l = 0; l < 3; ++l) {
            k_spmm_wmma<<<blocks, 128, 0, stream>>>(aptr, accol, acval, cur, nxt,
                                                    acc, NN);
            float* t = cur; cur = nxt; nxt = t;
        }
    }

    // 4) user-user part: ulat = uE; uacc = uE (reuse cur/nxt buffers)
    {
        int n4 = USER_N * LATDIM / 4;
        k_copy2<<<(n4 + 255) / 256, 256, 0, stream>>>((const float4*)uE,
                                                      (float4*)cur, (float4*)uacc, n4);
    }

    // 5) CSR for user-user adjacency
    k_zero_i32<<<(USER_N + 1 + 255) / 256, 256, 0, stream>>>(ucur, USER_N + 1);
    k_count<<<(UE + 255) / 256, 256, 0, stream>>>(uu_r, ucur, UE);
    k_scan<<<1, 1024, 0, stream>>>(ucur, uptr, ucur, USER_N);
    k_scatter<<<(UE + 255) / 256, 256, 0, stream>>>(uu_r, uu_c, uu_v, ucur,
                                                    uccol, ucval, UE);

    // 6) 2 user-user layers
    {
        int ntiles = (USER_N + 15) / 16;
        int blocks = (ntiles + 3) / 4;
        for (int l = 0; l < 2; ++l) {
            k_spmm_wmma<<<blocks, 128, 0, stream>>>(uptr, uccol, ucval, cur, nxt,
                                                    uacc, USER_N);
            float* t = cur; cur = nxt; nxt = t;
        }
    }
}